// GaussianScene2_76562087019173
// MI455X (gfx1250) — compile-verified
//
#include <hip/hip_runtime.h>
#include <math.h>

// Gaussian-splat preprocessing for MI455X (gfx1250, wave32).
// Bandwidth-bound (~1.7 FLOP/B): single fused pass, FP32 everywhere.
// The two 4x4 projections (ph@extr, ph@extr@intr) are fused into one
// 16x4 A-matrix and executed on the matrix pipe via V_WMMA_F32_16X16X4_F32
// (two WMMAs per wave cover 32 points; lanes decode pc from D[0..3] and
// pn from D[4..7] using the documented 16x16 f32 D layout).

typedef __attribute__((ext_vector_type(2))) float v2f;
typedef __attribute__((ext_vector_type(8))) float v8f;

#define TPB 128

__global__ __launch_bounds__(TPB) void gs_preprocess(
    const float* __restrict__ pts,     // (N,3)
    const float* __restrict__ scales,  // (N,3)
    const float* __restrict__ quats,   // (N,4)
    const float* __restrict__ cols,    // (N,3)
    const float* __restrict__ opac,    // (N,)
    const float* __restrict__ extr,    // 4x4 row-major
    const float* __restrict__ intr,    // 4x4 row-major
    const int* __restrict__ p_fx, const int* __restrict__ p_fy,
    const int* __restrict__ p_w,  const int* __restrict__ p_h,
    const int* __restrict__ p_ts,
    float* __restrict__ out, int N)
{
  const int tid  = blockIdx.x * TPB + (int)threadIdx.x;
  const int lane = (int)threadIdx.x & 31;
  const int base = (tid >> 5) << 5;        // wave's first gaussian (wave-uniform)
  if (base >= N) return;
  const int i = base + lane;

  const float fx = (float)p_fx[0];
  const float fy = (float)p_fy[0];
  const int W = p_w[0], H = p_h[0], TS = p_ts[0];

  float pc0, pc1, pc2, pn0, pn1, pn2, pn3;

  if (base + 32 <= N) {
    // ---------- WMMA point transform: D = A(16x4) x B(4x16) ----------
    // A rows 0-3 = extr^T, rows 4-7 = (extr@intr)^T, rows 8-15 repeat,
    // so both lane halves of the D layout decode the same way.
    const int mrow = lane & 15;
    const int row  = mrow & 7;               // 0..7
    const int k0   = (lane < 16) ? 0 : 2;    // A layout: vgpr0 holds K=0|2, vgpr1 K=1|3
    float a0, a1;
    if (row < 4) {                           // extr^T: A[m][k] = extr[k][m]
      a0 = extr[k0 * 4 + row];
      a1 = extr[(k0 + 1) * 4 + row];
    } else {                                 // (extr@intr)^T: A[4+m][k] = sum_t extr[k][t]*intr[t][m]
      const int m = row - 4;
      a0 = 0.0f; a1 = 0.0f;
      #pragma unroll
      for (int t = 0; t < 4; ++t) {
        a0 += extr[k0 * 4 + t] * intr[t * 4 + m];
        a1 += extr[(k0 + 1) * 4 + t] * intr[t * 4 + m];
      }
    }
    v2f A; A[0] = a0; A[1] = a1;

    // B layout: lanes 0-15 -> col N=lane, rows K=0(vgpr0),1(vgpr1);
    //           lanes 16-31 -> col N=lane-16, rows K=2(vgpr0),3(vgpr1).
    // Homogeneous point: (x, y, z, 1).
    const int plo = base + mrow;
    const int phi = plo + 16;
    v2f Blo, Bhi;
    if (lane < 16) {
      Blo[0] = pts[plo * 3 + 0]; Blo[1] = pts[plo * 3 + 1];
      Bhi[0] = pts[phi * 3 + 0]; Bhi[1] = pts[phi * 3 + 1];
    } else {
      Blo[0] = pts[plo * 3 + 2]; Blo[1] = 1.0f;
      Bhi[0] = pts[phi * 3 + 2]; Bhi[1] = 1.0f;
    }

    v8f z = {};
    // (neg_a, A, neg_b, B, c_mod, C, reuse_a, reuse_b)
    v8f d0 = __builtin_amdgcn_wmma_f32_16x16x4_f32(false, A, false, Blo, (short)0, z, false, false);
    v8f d1 = __builtin_amdgcn_wmma_f32_16x16x4_f32(false, A, false, Bhi, (short)0, z, false, false);

    // D layout: lane<16 -> rows 0-7 of D0 (cols = points base..base+15),
    //           lane>=16 -> rows 8-15 of D1 (cols = points base+16..base+31).
    const bool hi = lane >= 16;
    pc0 = hi ? d1[0] : d0[0];
    pc1 = hi ? d1[1] : d0[1];
    pc2 = hi ? d1[2] : d0[2];
    pn0 = hi ? d1[4] : d0[4];
    pn1 = hi ? d1[5] : d0[5];
    pn2 = hi ? d1[6] : d0[6];
    pn3 = hi ? d1[7] : d0[7];
  } else {
    // ---------- scalar fallback for a partial tail wave ----------
    if (i < N) {
      const float x = pts[i*3+0], y = pts[i*3+1], zz = pts[i*3+2];
      pc0 = x*extr[0] + y*extr[4] + zz*extr[ 8] + extr[12];
      pc1 = x*extr[1] + y*extr[5] + zz*extr[ 9] + extr[13];
      pc2 = x*extr[2] + y*extr[6] + zz*extr[10] + extr[14];
      const float pc3 = x*extr[3] + y*extr[7] + zz*extr[11] + extr[15];
      pn0 = pc0*intr[0] + pc1*intr[4] + pc2*intr[ 8] + pc3*intr[12];
      pn1 = pc0*intr[1] + pc1*intr[5] + pc2*intr[ 9] + pc3*intr[13];
      pn2 = pc0*intr[2] + pc1*intr[6] + pc2*intr[10] + pc3*intr[14];
      pn3 = pc0*intr[3] + pc1*intr[7] + pc2*intr[11] + pc3*intr[15];
    } else {
      pc0 = pc1 = 0.0f; pc2 = 1.0f;
      pn0 = pn1 = pn2 = 0.0f; pn3 = 1.0f;
    }
  }

  if (i >= N) return;

  // ---------- cov3d = (R*diag(s)) (R*diag(s))^T ----------
  const float s0 = scales[i*3+0], s1 = scales[i*3+1], s2 = scales[i*3+2];
  float qw = quats[i*4+0], qx = quats[i*4+1], qy = quats[i*4+2], qz = quats[i*4+3];
  const float qinv = 1.0f / sqrtf(qw*qw + qx*qx + qy*qy + qz*qz);
  qw *= qinv; qx *= qinv; qy *= qinv; qz *= qinv;
  const float m00 = (1.0f - 2.0f*(qy*qy + qz*qz)) * s0;
  const float m01 = (2.0f*(qx*qy - qw*qz)) * s1;
  const float m02 = (2.0f*(qx*qz + qw*qy)) * s2;
  const float m10 = (2.0f*(qx*qy + qw*qz)) * s0;
  const float m11 = (1.0f - 2.0f*(qx*qx + qz*qz)) * s1;
  const float m12 = (2.0f*(qy*qz - qw*qx)) * s2;
  const float m20 = (2.0f*(qx*qz - qw*qy)) * s0;
  const float m21 = (2.0f*(qy*qz + qw*qx)) * s1;
  const float m22 = (1.0f - 2.0f*(qx*qx + qy*qy)) * s2;

  float C[3][3];
  C[0][0] = m00*m00 + m01*m01 + m02*m02;
  C[0][1] = m00*m10 + m01*m11 + m02*m12;
  C[0][2] = m00*m20 + m01*m21 + m02*m22;
  C[1][1] = m10*m10 + m11*m11 + m12*m12;
  C[1][2] = m10*m20 + m11*m21 + m12*m22;
  C[2][2] = m20*m20 + m21*m21 + m22*m22;
  C[1][0] = C[0][1]; C[2][0] = C[0][2]; C[2][1] = C[1][2];

  // ---------- Mid = E^T C E  (E = extr[:3,:3]) ----------
  float E[3][3];
  #pragma unroll
  for (int r = 0; r < 3; ++r)
    #pragma unroll
    for (int c = 0; c < 3; ++c)
      E[r][c] = extr[r*4 + c];

  float T[3][3], Mid[3][3];
  #pragma unroll
  for (int j = 0; j < 3; ++j)
    #pragma unroll
    for (int l = 0; l < 3; ++l)
      T[j][l] = C[j][0]*E[0][l] + C[j][1]*E[1][l] + C[j][2]*E[2][l];
  #pragma unroll
  for (int r = 0; r < 3; ++r)
    #pragma unroll
    for (int l = 0; l < 3; ++l)
      Mid[r][l] = E[0][r]*T[0][l] + E[1][r]*T[1][l] + E[2][r]*T[2][l];

  // ---------- cov2d = J Mid J^T ----------
  const float zinv  = 1.0f / pc2;
  const float zinv2 = zinv * zinv;
  const float j00 = fx * zinv;
  const float j02 = fx * pc0 * zinv2;
  const float j11 = fy * zinv;
  const float j12 = fy * pc1 * zinv2;
  const float u0 = j00*Mid[0][0] + j02*Mid[2][0];
  const float u1 = j00*Mid[0][1] + j02*Mid[2][1];
  const float u2 = j00*Mid[0][2] + j02*Mid[2][2];
  const float v0 = j11*Mid[1][0] + j12*Mid[2][0];
  const float v1 = j11*Mid[1][1] + j12*Mid[2][1];
  const float v2 = j11*Mid[1][2] + j12*Mid[2][2];
  const float ca = u0*j00 + u2*j02;
  const float cb = u1*j11 + u2*j12;
  const float cc = v0*j00 + v2*j02;
  const float cd = v1*j11 + v2*j12;

  // ---------- NDC / mask / ellipse ----------
  const float ipn3 = 1.0f / pn3;
  const float nd0 = pn0 * ipn3, nd1 = pn1 * ipn3, nd2 = pn2 * ipn3;
  const bool mask = (nd2 > 0.2f) && (nd0 < 1.3f) && (nd0 > -1.3f) &&
                    (nd1 < 1.3f) && (nd1 > -1.3f);
  const float det  = ca*cd - cb*cc;
  const float idet = 1.0f / det;
  const float i00 =  cd*idet, i01 = -cb*idet, i10 = -cc*idet, i11 = ca*idet;
  const float mid = 0.5f*(ca + cd);
  const float lam = mid + sqrtf(fmaxf(mid*mid - det, 0.1f));
  const float radius = ceilf(3.0f * sqrtf(lam));
  const float px = ((nd0 + 1.0f)*(float)W - 1.0f)*0.5f;
  const float py = ((nd1 + 1.0f)*(float)H - 1.0f)*0.5f;

  const int max_tx = (W + TS - 1)/TS - 1;
  const int max_ty = (H + TS - 1)/TS - 1;
  const float fts = (float)TS;
  int tlx = (int)((px - radius)/fts); tlx = tlx < 0 ? 0 : (tlx > max_tx ? max_tx : tlx);
  int tly = (int)((py - radius)/fts); tly = tly < 0 ? 0 : (tly > max_ty ? max_ty : tly);
  int brx = (int)((px + radius)/fts); brx = brx < 0 ? 0 : (brx > max_tx ? max_tx : brx);
  int bry = (int)((py + radius)/fts); bry = bry < 0 ? 0 : (bry > max_ty ? max_ty : bry);
  int spanx = brx + 1 - tlx; spanx = spanx < 1 ? 1 : spanx;
  int spany = bry + 1 - tly; spany = spany < 1 ? 1 : spany;
  const int tiles = mask ? spanx * spany : 0;
  const float mf = mask ? 1.0f : 0.0f;

  // ---------- stores (outputs concatenated flat, return order) ----------
  const long long Nl = (long long)N;
  float* means_o = out;
  float* cov_o   = out + 3*Nl;
  float* icov_o  = out + 7*Nl;
  float* rad_o   = out + 11*Nl;
  float* tiles_o = out + 12*Nl;
  float* tlx_o   = out + 13*Nl;
  float* tly_o   = out + 14*Nl;
  float* brx_o   = out + 15*Nl;
  float* bry_o   = out + 16*Nl;
  float* col_o   = out + 17*Nl;
  float* op_o    = out + 20*Nl;
  float* mask_o  = out + 21*Nl;

  means_o[i*3+0] = px*mf;  means_o[i*3+1] = py*mf;  means_o[i*3+2] = nd2*mf;
  cov_o[i*4+0] = ca*mf;  cov_o[i*4+1] = cb*mf;  cov_o[i*4+2] = cc*mf;  cov_o[i*4+3] = cd*mf;
  icov_o[i*4+0] = i00*mf; icov_o[i*4+1] = i01*mf; icov_o[i*4+2] = i10*mf; icov_o[i*4+3] = i11*mf;
  rad_o[i]   = radius*mf;
  tiles_o[i] = (float)tiles;
  tlx_o[i] = (float)tlx; tly_o[i] = (float)tly;
  brx_o[i] = (float)brx; bry_o[i] = (float)bry;
  col_o[i*3+0] = cols[i*3+0]*mf; col_o[i*3+1] = cols[i*3+1]*mf; col_o[i*3+2] = cols[i*3+2]*mf;
  op_o[i]   = opac[i]*mf;
  mask_o[i] = mf;
}

extern "C" void kernel_launch(void* const* d_in, const int* in_sizes, int n_in,
                              void* d_out, int out_size, void* d_ws, size_t ws_size,
                              hipStream_t stream) {
  (void)n_in; (void)out_size; (void)d_ws; (void)ws_size;
  const float* pts    = (const float*)d_in[0];
  const float* scales = (const float*)d_in[1];
  const float* quats  = (const float*)d_in[2];
  const float* cols   = (const float*)d_in[3];
  const float* opac   = (const float*)d_in[4];
  const float* extr   = (const float*)d_in[5];
  const float* intr   = (const float*)d_in[6];
  const int* fx = (const int*)d_in[7];
  const int* fy = (const int*)d_in[8];
  const int* w  = (const int*)d_in[9];
  const int* h  = (const int*)d_in[10];
  const int* ts = (const int*)d_in[11];
  const int N = in_sizes[4];             // opacity length == N
  const int blocks = (N + TPB - 1) / TPB;
  gs_preprocess<<<blocks, TPB, 0, stream>>>(pts, scales, quats, cols, opac,
                                            extr, intr, fx, fy, w, h, ts,
                                            (float*)d_out, N);
}